// TemperalAttention_55430847922846
// MI455X (gfx1250) — compile-verified
//
#include <hip/hip_runtime.h>
#include <math.h>
#include <stdint.h>

typedef __attribute__((ext_vector_type(2))) float v2f;
typedef __attribute__((ext_vector_type(8))) float v8f;

#define BDIM 256
#define PDIM 4096
#define MDIM 64
#define SPLITS 8
#define TPAD 68   // padded LDS tile row stride in floats (bank-conflict-free)
#define CHUNKS_PER_WAVE (PDIM / SPLITS / 8 / 16)   // 4

// branch-free tanh: saturates correctly at +/-1 via v_exp/v_rcp
__device__ __forceinline__ float fast_tanh(float x) {
    float e = __expf(2.0f * x);
    return 1.0f - 2.0f * __builtin_amdgcn_rcpf(e + 1.0f);
}

// ---- stage W_uh + fused ds bias + W_l into LDS (per block) ----
__device__ __forceinline__ void stage_block(
    int b, int tid,
    const float* __restrict__ hidden, const float* __restrict__ cell,
    const float* __restrict__ W_ds, const float* __restrict__ b_ds,
    const float* __restrict__ W_uh, const float* __restrict__ b_uh,
    const float* __restrict__ W_l,
    float* sWuh, float* sDsb, float* sWl)
{
    for (int i = tid; i < MDIM * MDIM; i += BDIM) sWuh[i] = W_uh[i];
    if (tid < MDIM) {
        sWl[tid] = W_l[tid];
        float acc = b_ds[tid] + b_uh[tid];
        const float* h = hidden + b * MDIM;
        const float* c = cell   + b * MDIM;
        #pragma unroll 8
        for (int f = 0; f < MDIM; ++f) acc += h[f] * W_ds[f * MDIM + tid];
        #pragma unroll 8
        for (int f = 0; f < MDIM; ++f) acc += c[f] * W_ds[(MDIM + f) * MDIM + tid];
        sDsb[tid] = acc;
    }
}

// ---- preload B fragments (W_uh) + epilogue constants into registers ----
__device__ __forceinline__ void load_bfrags(
    const float* sWuh, const float* sDsb, const float* sWl,
    v2f (&bfr)[4][16], float (&wlv)[4], float (&dsv)[4])
{
    const int lane = threadIdx.x & 31;
    const int rowInTile = lane & 15;
    const int halfsel = lane >> 4;
    #pragma unroll
    for (int t = 0; t < 4; ++t) {
        const int mcol = t * 16 + rowInTile;
        const float* wcol = sWuh + mcol;
        #pragma unroll
        for (int kk = 0; kk < 16; ++kk) {
            const int k0 = kk * 4 + 2 * halfsel;
            bfr[t][kk].x = wcol[(k0    ) * MDIM];
            bfr[t][kk].y = wcol[(k0 + 1) * MDIM];
        }
        wlv[t] = sWl[mcol];
        dsv[t] = sDsb[mcol];
    }
}

// ---- core: 64 WMMAs + fused tanh/dot epilogue + lane reduce + store ----
__device__ __forceinline__ void chunk16_mm(
    const v2f (&afr)[16], const v2f (&bfr)[4][16],
    const float (&wlv)[4], const float (&dsv)[4],
    float bl, float* dst)
{
    const int lane = threadIdx.x & 31;
    const int rowInTile = lane & 15;
    const int halfsel = lane >> 4;

    float pl[8];
    #pragma unroll
    for (int v = 0; v < 8; ++v) pl[v] = 0.0f;

    #pragma unroll
    for (int t = 0; t < 4; ++t) {
        v8f c = {};
        #pragma unroll
        for (int kk = 0; kk < 16; ++kk)
            c = __builtin_amdgcn_wmma_f32_16x16x4_f32(
                    false, afr[kk], false, bfr[t][kk], (short)0, c, false, false);
        #pragma unroll
        for (int v = 0; v < 8; ++v)
            pl[v] += fast_tanh(c[v] + dsv[t]) * wlv[t];
    }

    #pragma unroll
    for (int m = 1; m < 16; m <<= 1) {
        #pragma unroll
        for (int v = 0; v < 8; ++v)
            pl[v] += __shfl_xor(pl[v], m, 32);
    }
    if (rowInTile == 0) {
        #pragma unroll
        for (int v = 0; v < 8; ++v)
            dst[halfsel * 8 + v] = pl[v] + bl;
    }
}

// ---- issue 8 coalesced async global->LDS b128 loads for one 16x64 tile ----
__device__ __forceinline__ void issue_tile_loads(
    const float* __restrict__ Xb, int p0, float* tile)
{
    const int lane = threadIdx.x & 31;
    unsigned ladr = (unsigned)(uintptr_t)tile
                  + (unsigned)((lane >> 4) * TPAD * 4 + (lane & 15) * 16);
    unsigned gofs = (unsigned)p0 * (MDIM * 4) + (unsigned)lane * 16;
    #pragma unroll
    for (int i = 0; i < 8; ++i) {
        asm volatile("global_load_async_to_lds_b128 %0, %1, %2"
                     :: "v"(ladr), "v"(gofs), "s"(Xb) : "memory");
        ladr += 2 * TPAD * 4;
        gofs += 512;
    }
}

// ---- read A fragments from padded LDS tile and compute ----
__device__ __forceinline__ void compute_tile(
    const float* tile,
    const v2f (&bfr)[4][16], const float (&wlv)[4], const float (&dsv)[4],
    float bl, float* dst)
{
    const int lane = threadIdx.x & 31;
    const int rowInTile = lane & 15;
    const int halfsel = lane >> 4;
    v2f afr[16];
    const float* trow = tile + rowInTile * TPAD + halfsel * 2;
    #pragma unroll
    for (int kk = 0; kk < 16; ++kk) {
        afr[kk].x = trow[kk * 4 + 0];
        afr[kk].y = trow[kk * 4 + 1];
    }
    chunk16_mm(afr, bfr, wlv, dsv, bl, dst);
}

// ---- direct path (fallback kernel): per-lane global b64 A loads ----
__device__ __forceinline__ void chunk16_direct(
    const float* __restrict__ Xb, int p0,
    const v2f (&bfr)[4][16], const float (&wlv)[4], const float (&dsv)[4],
    float bl, float* dst)
{
    const int lane = threadIdx.x & 31;
    const int rowInTile = lane & 15;
    const int halfsel = lane >> 4;
    v2f afr[16];
    const float* xrow = Xb + (size_t)(p0 + rowInTile) * MDIM + halfsel * 2;
    #pragma unroll
    for (int kk = 0; kk < 16; ++kk) {
        afr[kk].x = xrow[kk * 4 + 0];
        afr[kk].y = xrow[kk * 4 + 1];
    }
    chunk16_mm(afr, bfr, wlv, dsv, bl, dst);
}

// ================= phase 1: logits, grid (B, SPLITS) =================
__global__ __launch_bounds__(BDIM) void logits_kernel(
    const float* __restrict__ hidden, const float* __restrict__ cell,
    const float* __restrict__ X, const float* __restrict__ W_ds,
    const float* __restrict__ b_ds, const float* __restrict__ W_uh,
    const float* __restrict__ b_uh, const float* __restrict__ W_l,
    const float* __restrict__ b_l, float* __restrict__ ws)
{
    __shared__ float sWuh[MDIM * MDIM];
    __shared__ float sDsb[MDIM];
    __shared__ float sWl[MDIM];
    __shared__ float sTile[8][2][16 * TPAD];   // double-buffered per wave

    const int b   = blockIdx.x;
    const int s   = blockIdx.y;
    const int tid = threadIdx.x;
    const int wave = tid >> 5;

    stage_block(b, tid, hidden, cell, W_ds, b_ds, W_uh, b_uh, W_l, sWuh, sDsb, sWl);
    __syncthreads();

    v2f bfr[4][16]; float wlv[4], dsv[4];
    load_bfrags(sWuh, sDsb, sWl, bfr, wlv, dsv);

    const float bl = b_l[0];
    const float* Xb = X + (size_t)b * PDIM * MDIM;
    float* wsb = ws + (size_t)b * PDIM;

    const int base = s * (PDIM / SPLITS) + wave * (CHUNKS_PER_WAVE * 16);

    // double-buffered pipeline: keep one tile's loads in flight while computing
    issue_tile_loads(Xb, base, sTile[wave][0]);
    #pragma unroll
    for (int ci = 0; ci < CHUNKS_PER_WAVE; ++ci) {
        const int p0 = base + ci * 16;
        if (ci + 1 < CHUNKS_PER_WAVE) {
            // buffer being refilled was last read by the previous chunk -> drain ds reads
            asm volatile("s_wait_dscnt 0x0" ::: "memory");
            issue_tile_loads(Xb, p0 + 16, sTile[wave][(ci + 1) & 1]);
            asm volatile("s_wait_asynccnt 0x8" ::: "memory");  // current tile ready
        } else {
            asm volatile("s_wait_asynccnt 0x0" ::: "memory");
        }
        compute_tile(sTile[wave][ci & 1], bfr, wlv, dsv, bl, wsb + p0);
    }
}

// ================= phase 2: softmax over P per batch =================
__global__ __launch_bounds__(BDIM) void softmax_kernel(
    const float* __restrict__ ws, float* __restrict__ out)
{
    __shared__ float redA[8], redB[8];
    const int b = blockIdx.x;
    const int tid = threadIdx.x;
    const int lane = tid & 31;
    const int wave = tid >> 5;
    const float* wb = ws + (size_t)b * PDIM;

    float v[16];
    float lmax = -INFINITY;
    #pragma unroll
    for (int i = 0; i < 16; ++i) {
        v[i] = wb[tid + i * BDIM];
        lmax = fmaxf(lmax, v[i]);
    }
    #pragma unroll
    for (int m = 16; m >= 1; m >>= 1) lmax = fmaxf(lmax, __shfl_xor(lmax, m, 32));
    if (lane == 0) redA[wave] = lmax;
    __syncthreads();
    float gmax = redA[0];
    #pragma unroll
    for (int i = 1; i < 8; ++i) gmax = fmaxf(gmax, redA[i]);

    float psum = 0.0f;
    #pragma unroll
    for (int i = 0; i < 16; ++i) { v[i] = __expf(v[i] - gmax); psum += v[i]; }
    #pragma unroll
    for (int m = 16; m >= 1; m >>= 1) psum += __shfl_xor(psum, m, 32);
    if (lane == 0) redB[wave] = psum;
    __syncthreads();
    float gsum = 0.0f;
    #pragma unroll
    for (int i = 0; i < 8; ++i) gsum += redB[i];
    const float inv = 1.0f / gsum;

    float* ob = out + (size_t)b * PDIM;
    #pragma unroll
    for (int i = 0; i < 16; ++i) ob[tid + i * BDIM] = v[i] * inv;
}

// ============ fallback: fused one-block-per-b (small ws) ============
__global__ __launch_bounds__(BDIM) void fused_kernel(
    const float* __restrict__ hidden, const float* __restrict__ cell,
    const float* __restrict__ X, const float* __restrict__ W_ds,
    const float* __restrict__ b_ds, const float* __restrict__ W_uh,
    const float* __restrict__ b_uh, const float* __restrict__ W_l,
    const float* __restrict__ b_l, float* __restrict__ out)
{
    __shared__ float sWuh[MDIM * MDIM];
    __shared__ float sDsb[MDIM];
    __shared__ float sWl[MDIM];
    __shared__ float sL[PDIM];
    __shared__ float redA[8], redB[8];

    const int b = blockIdx.x;
    const int tid = threadIdx.x;
    const int lane = tid & 31;
    const int wave = tid >> 5;

    stage_block(b, tid, hidden, cell, W_ds, b_ds, W_uh, b_uh, W_l, sWuh, sDsb, sWl);
    __syncthreads();

    v2f bfr[4][16]; float wlv[4], dsv[4];
    load_bfrags(sWuh, sDsb, sWl, bfr, wlv, dsv);

    const float bl = b_l[0];
    const float* Xb = X + (size_t)b * PDIM * MDIM;

    for (int p0 = wave * 512; p0 < wave * 512 + 512; p0 += 16)
        chunk16_direct(Xb, p0, bfr, wlv, dsv, bl, &sL[p0]);
    __syncthreads();

    float lmax = -INFINITY;
    for (int i = tid; i < PDIM; i += BDIM) lmax = fmaxf(lmax, sL[i]);
    #pragma unroll
    for (int m = 16; m >= 1; m >>= 1) lmax = fmaxf(lmax, __shfl_xor(lmax, m, 32));
    if (lane == 0) redA[wave] = lmax;
    __syncthreads();
    float gmax = redA[0];
    #pragma unroll
    for (int i = 1; i < 8; ++i) gmax = fmaxf(gmax, redA[i]);

    float psum = 0.0f;
    for (int i = tid; i < PDIM; i += BDIM) {
        float e = __expf(sL[i] - gmax);
        sL[i] = e;
        psum += e;
    }
    #pragma unroll
    for (int m = 16; m >= 1; m >>= 1) psum += __shfl_xor(psum, m, 32);
    if (lane == 0) redB[wave] = psum;
    __syncthreads();
    float gsum = 0.0f;
    #pragma unroll
    for (int i = 0; i < 8; ++i) gsum += redB[i];
    const float inv = 1.0f / gsum;
    float* ob = out + (size_t)b * PDIM;
    for (int i = tid; i < PDIM; i += BDIM) ob[i] = sL[i] * inv;
}

extern "C" void kernel_launch(void* const* d_in, const int* in_sizes, int n_in,
                              void* d_out, int out_size, void* d_ws, size_t ws_size,
                              hipStream_t stream) {
    const float* hidden = (const float*)d_in[0];
    const float* cell   = (const float*)d_in[1];
    const float* X      = (const float*)d_in[2];
    const float* W_ds   = (const float*)d_in[3];
    const float* b_ds   = (const float*)d_in[4];
    const float* W_uh   = (const float*)d_in[5];
    const float* b_uh   = (const float*)d_in[6];
    const float* W_l    = (const float*)d_in[7];
    const float* b_l    = (const float*)d_in[8];
    float* out = (float*)d_out;

    const int B = in_sizes[0] / MDIM;   // 128
    const size_t need = (size_t)B * PDIM * sizeof(float);
    if (ws_size >= need) {
        logits_kernel<<<dim3(B, SPLITS), BDIM, 0, stream>>>(
            hidden, cell, X, W_ds, b_ds, W_uh, b_uh, W_l, b_l, (float*)d_ws);
        softmax_kernel<<<B, BDIM, 0, stream>>>((const float*)d_ws, out);
    } else {
        fused_kernel<<<B, BDIM, 0, stream>>>(
            hidden, cell, X, W_ds, b_ds, W_uh, b_uh, W_l, b_l, out);
    }
    (void)n_in; (void)out_size;
}